// MVGRLEncoder_44839458570832
// MI455X (gfx1250) — compile-verified
//
#include <hip/hip_runtime.h>

typedef __attribute__((ext_vector_type(2))) float v2f;
typedef __attribute__((ext_vector_type(8))) float v8f;

#define NN 40000
#define EE 640000
#define DD 128
#define GG 128

// ---------------------------------------------------------------------------
// GEMM: Y = X @ W^T   (X: n x 128 row-major, W: 128 x 128 row-major)
// One wave computes a 16x128 output strip using V_WMMA_F32_16X16X4_F32.
// A-frag layout (16x4 f32): lanes 0-15 -> M=lane, K = {0,1}; lanes 16-31 -> M=lane-16, K = {2,3}
// B-frag layout (4x16 f32, mirrored):  lanes 0-15 -> N=lane, K = {0,1}; lanes 16-31 -> N=lane-16, K={2,3}
// C/D (16x16 f32): vgpr v -> row v + 8*(lane>=16), col = lane&15
// ---------------------------------------------------------------------------
__global__ __launch_bounds__(128) void gemm_xwt_wmma(const float* __restrict__ X,
                                                     const float* __restrict__ W,
                                                     float* __restrict__ Y) {
    __shared__ float Wlds[DD * DD];
    // cooperative load of W into LDS (64 KB), float4 granularity
    for (int i = threadIdx.x; i < DD * DD / 4; i += blockDim.x) {
        ((float4*)Wlds)[i] = ((const float4*)W)[i];
    }
    __syncthreads();

    const int wave = threadIdx.x >> 5;
    const int lane = threadIdx.x & 31;
    const int m0   = (blockIdx.x * 4 + wave) * 16;     // 2500 M-tiles total
    const int mrow = m0 + (lane & 15);
    const int khalf = (lane >> 4) * 2;                 // 0 or 2
    const int nlo  = lane & 15;

    v8f c[8] = {};                                     // 8 N-tiles of 16x16 f32 acc

    const float* xrow = X + (size_t)mrow * DD;
    for (int k0 = 0; k0 < DD; k0 += 4) {
        v2f a;
        a.x = xrow[k0 + khalf + 0];
        a.y = xrow[k0 + khalf + 1];
#pragma unroll
        for (int nt = 0; nt < 8; ++nt) {
            const int n = nt * 16 + nlo;
            v2f b;
            b.x = Wlds[n * DD + k0 + khalf + 0];
            b.y = Wlds[n * DD + k0 + khalf + 1];
            c[nt] = __builtin_amdgcn_wmma_f32_16x16x4_f32(
                false, a, false, b, (short)0, c[nt], false, false);
        }
    }

    const int rbase = m0 + (lane >> 4) * 8;
#pragma unroll
    for (int nt = 0; nt < 8; ++nt) {
#pragma unroll
        for (int v = 0; v < 8; ++v) {
            Y[(size_t)(rbase + v) * DD + nt * 16 + nlo] = c[nt][v];
        }
    }
}

// ---------------------------------------------------------------------------
// Edge scatter: acc[col[e]] += lin[row[e]] * ew[e]
// One wave per edge; lane handles 4 consecutive floats (float4 gather, 4 fp atomics).
// acc (20.5 MB) is L2-resident -> atomics resolve in L2.
// ---------------------------------------------------------------------------
__global__ __launch_bounds__(256) void edge_scatter(const float* __restrict__ lin,
                                                    const int* __restrict__ row,
                                                    const int* __restrict__ col,
                                                    const float* __restrict__ ew,
                                                    float* __restrict__ acc) {
    const int e    = (blockIdx.x * blockDim.x + threadIdx.x) >> 5;
    const int lane = threadIdx.x & 31;
    if (e >= EE) return;
    const int r = row[e];
    const int cidx = col[e];
    const float w = ew[e];
    const float4 v = ((const float4*)(lin + (size_t)r * DD))[lane];
    float* dst = acc + (size_t)cidx * DD + lane * 4;
    unsafeAtomicAdd(dst + 0, v.x * w);
    unsafeAtomicAdd(dst + 1, v.y * w);
    unsafeAtomicAdd(dst + 2, v.z * w);
    unsafeAtomicAdd(dst + 3, v.w * w);
}

// ---------------------------------------------------------------------------
// PReLU + graph pooling: y = prelu(acc); hout = y; hg[batch[n]] += y
// One thread per (node, feature); hg stride is 3*D (concat layout).
// ---------------------------------------------------------------------------
__global__ __launch_bounds__(256) void prelu_pool(const float* __restrict__ acc,
                                                  const float* __restrict__ aslope,
                                                  const int* __restrict__ batch,
                                                  float* __restrict__ hout,
                                                  float* __restrict__ hg) {
    const int idx  = blockIdx.x * blockDim.x + threadIdx.x;
    const int node = idx >> 7;
    const int d    = idx & 127;
    if (node >= NN) return;
    const float alpha = aslope[0];
    const float x = acc[(size_t)node * DD + d];
    const float y = (x >= 0.0f) ? x : alpha * x;
    hout[(size_t)node * DD + d] = y;
    const int g = batch[node];
    unsafeAtomicAdd(hg + (size_t)g * (3 * DD) + d, y);
}

// ---------------------------------------------------------------------------
// Driver
// inputs: feat, edge_weight, W0, W1, W2, a0, a1, a2, edge_index, batch_indices
// d_out: [ h_final (40000x128) | hg (128x384) ]
// d_ws : two N*D f32 ping-pong buffers (40.96 MB)
// ---------------------------------------------------------------------------
extern "C" void kernel_launch(void* const* d_in, const int* in_sizes, int n_in,
                              void* d_out, int out_size, void* d_ws, size_t ws_size,
                              hipStream_t stream) {
    const float* feat  = (const float*)d_in[0];
    const float* ew    = (const float*)d_in[1];
    const float* Ws[3] = {(const float*)d_in[2], (const float*)d_in[3], (const float*)d_in[4]};
    const float* as[3] = {(const float*)d_in[5], (const float*)d_in[6], (const float*)d_in[7]};
    const int*   eidx  = (const int*)d_in[8];
    const int*   row   = eidx;
    const int*   col   = eidx + EE;
    const int*   batch = (const int*)d_in[9];

    float* out = (float*)d_out;
    float* hg  = out + (size_t)NN * DD;          // 128 x 384
    float* bufs[2] = {(float*)d_ws, (float*)d_ws + (size_t)NN * DD};

    hipMemsetAsync(hg, 0, (size_t)GG * 3 * DD * sizeof(float), stream);

    const float* hin = feat;
    for (int L = 0; L < 3; ++L) {
        float* lin = bufs[L & 1];
        float* acc = bufs[(L + 1) & 1];

        gemm_xwt_wmma<<<NN / 16 / 4, 128, 0, stream>>>(hin, Ws[L], lin);
        hipMemsetAsync(acc, 0, (size_t)NN * DD * sizeof(float), stream);
        edge_scatter<<<EE * 32 / 256, 256, 0, stream>>>(lin, row, col, ew, acc);

        float* hout = (L == 2) ? out : lin;      // lin buffer is free after scatter
        prelu_pool<<<NN * DD / 256, 256, 0, stream>>>(acc, as[L], batch, hout, hg + L * DD);
        hin = hout;
    }
}